// UpSamplingWithArgmax2D_85839216378554
// MI455X (gfx1250) — compile-verified
//
#include <hip/hip_runtime.h>

// UpSamplingWithArgmax2D (max-unpool) for MI455X / gfx1250.
//
// Shapes (fixed by the reference):
//   x:       [B=16, H=128, W=128, C=64]  float32
//   indices: [B,H,W,C] int32, per-batch flat index into (Hout*Wout*C),
//            guaranteed to land in element's own 2x2 window -> collision-free.
//   out:     [B, Hout=256, Wout=256, C=64] float32
//
// Strategy: pure HBM-streaming kernel (0 FLOPs -> WMMA inapplicable).
//   384 MiB total traffic @ 23.3 TB/s  => ~17 us roofline.
//   - 1 thread = 4 consecutive channels of one (b,h,w) window.
//   - 2x 128-bit NT loads (x4, idx4), 4x 128-bit NT stores (the 2x2 window).
//   - Non-temporal cache policy: streams are touched once; output (256 MiB)
//     exceeds the 192 MB L2, so keep it out of the cache (th:TH_*_NT).
//   - global_prefetch_b8 on the x stream (gfx1250 prefetch path).

typedef __attribute__((ext_vector_type(4))) float v4f;
typedef __attribute__((ext_vector_type(4))) int   v4i;

namespace {
constexpr int B = 16, H = 128, W = 128, C = 64;
constexpr int HOUT = 2 * H, WOUT = 2 * W;
constexpr int N4 = B * H * W * (C / 4);            // 4,194,304 threads
constexpr int ROW_STRIDE = WOUT * C;               // elements per output row
constexpr long long OUT_PER_BATCH = (long long)HOUT * WOUT * C;
constexpr int BLOCK = 256;                         // 8 wave32 waves
}  // namespace

__global__ __launch_bounds__(BLOCK) void
unpool_argmax2d_kernel(const float* __restrict__ x,
                       const int* __restrict__ indices,
                       float* __restrict__ out) {
  const int t = blockIdx.x * BLOCK + threadIdx.x;   // grid sized exactly: no tail

  // Decode (b,h,w,c4) from t; linearization matches [B,H,W,C] layout, so the
  // flat float4 input offset is simply t.
  const int c = (t & 15) << 2;          // channel base, multiple of 4
  const int w = (t >> 4) & (W - 1);
  const int h = (t >> 11) & (H - 1);
  const int b = t >> 18;

  // Prefetch the x stream ~128 KiB ahead (clamped in-range) -> global_prefetch_b8.
  {
    const int pf = t + 8192 < N4 ? t + 8192 : N4 - 1;
    __builtin_prefetch((const void*)((const v4f*)x + pf), 0, 1);
  }

  const v4f xv = __builtin_nontemporal_load((const v4f*)x + t);
  const v4i iv = __builtin_nontemporal_load((const v4i*)indices + t);

  // Within-batch flat offset of window position (dy=0,dx=0) for channel base c.
  const int base = (2 * h) * ROW_STRIDE + (2 * w) * C + c;
  float* __restrict__ outb = out + (long long)b * OUT_PER_BATCH + base;

  // Emit one 128-bit NT store per window position; component j carries x_j iff
  // that channel's argmax index targets this position, else 0 (collision-free).
#pragma unroll
  for (int dy = 0; dy < 2; ++dy) {
#pragma unroll
    for (int dx = 0; dx < 2; ++dx) {
      const int off = dy * ROW_STRIDE + dx * C;
      const int tgt = base + off;
      v4f v;
      v.x = (iv.x == tgt + 0) ? xv.x : 0.0f;
      v.y = (iv.y == tgt + 1) ? xv.y : 0.0f;
      v.z = (iv.z == tgt + 2) ? xv.z : 0.0f;
      v.w = (iv.w == tgt + 3) ? xv.w : 0.0f;
      __builtin_nontemporal_store(v, (v4f*)(outb + off));
    }
  }
}

extern "C" void kernel_launch(void* const* d_in, const int* in_sizes, int n_in,
                              void* d_out, int out_size, void* d_ws, size_t ws_size,
                              hipStream_t stream) {
  (void)in_sizes; (void)n_in; (void)out_size; (void)d_ws; (void)ws_size;
  const float* x       = (const float*)d_in[0];
  const int*   indices = (const int*)d_in[1];
  float*       out     = (float*)d_out;

  const int grid = N4 / BLOCK;  // 16384 blocks, exact cover (every output written)
  unpool_argmax2d_kernel<<<grid, BLOCK, 0, stream>>>(x, indices, out);
}